// WaveletFeatureAugmentation_60705067761823
// MI455X (gfx1250) — compile-verified
//
#include <hip/hip_runtime.h>

typedef __attribute__((ext_vector_type(2))) float v2f;
typedef __attribute__((ext_vector_type(8))) float v8f;

#define ROW_IN  4096
#define ROW_OUT 4110
#define NROWS   (32 * 512)   // 16384

// pywt db4 dec_lo reversed (correlation kernel), KREV[k] = dec_lo[7-k]
__device__ __constant__ float KREV[8] = {
     0.23037781330885523f,  0.7148465705525415f,   0.6308807679295904f,
    -0.02798376941698385f, -0.18703481171888114f,  0.030841381835986965f,
     0.032883011666982945f, -0.010597401784997278f
};

// Level lengths: n_{l+1} = floor((n_l + 7)/2)
__device__ __constant__ int NS[10] = {4096, 2051, 1029, 518, 262, 134, 70, 38, 22, 14};

// Symmetric reflection (pywt 'symmetric'): [-n, 2n) -> [0, n)
__device__ __forceinline__ int symref(int m, int n) {
    m = (m < 0)  ? (-1 - m)        : m;
    m = (m >= n) ? (2 * n - 1 - m) : m;
    return m;
}

// ---------------------------------------------------------------------------
// Kernel 1: per-row stream copy x -> out[0:4096] + 9-level DWT in LDS,
// approx (14 floats) parked at out[row*4110 + 4096 .. +4109].
// ---------------------------------------------------------------------------
__global__ __launch_bounds__(256) void k_copy_dwt(const float* __restrict__ x,
                                                  float* __restrict__ out) {
    __shared__ float bufA[4096];
    __shared__ float bufB[2051];

    const int r = blockIdx.x;
    const int t = threadIdx.x;
    const float* __restrict__ xr   = x   + (size_t)r * ROW_IN;
    float*       __restrict__ outr = out + (size_t)r * ROW_OUT;

    // 16 floats per thread: coalesced float4 loads (row base 16KB-aligned),
    // float2 stores to out (out row base only 8B-aligned on odd rows).
    #pragma unroll
    for (int v = 0; v < 4; ++v) {
        const int q = v * 256 + t;      // float4 index within row
        float4 d = ((const float4*)xr)[q];
        const int e = q * 4;
        bufA[e + 0] = d.x; bufA[e + 1] = d.y;
        bufA[e + 2] = d.z; bufA[e + 3] = d.w;
        float2* o2 = (float2*)(outr + e);
        o2[0] = make_float2(d.x, d.y);
        o2[1] = make_float2(d.z, d.w);
    }
    __syncthreads();

    float* src = bufA;
    float* dst = bufB;
    #pragma unroll 1
    for (int lvl = 0; lvl < 9; ++lvl) {
        const int nin  = NS[lvl];
        const int nout = NS[lvl + 1];
        for (int i = t; i < nout; i += 256) {
            const int base = 2 * i - 6;   // left pad of 6 after pywt's [1:] slice
            float acc = 0.0f;
            #pragma unroll
            for (int k = 0; k < 8; ++k)
                acc += src[symref(base + k, nin)] * KREV[k];
            dst[i] = acc;
        }
        __syncthreads();
        float* tmp = src; src = dst; dst = tmp;
    }

    // approx now in src[0..13]
    if (t < 14) outr[ROW_IN + t] = src[t];
}

// ---------------------------------------------------------------------------
// Kernel 2: aug = approx @ W^T + bias via V_WMMA_F32_16X16X4_F32.
// Each wave32 owns a 16-row tile; K=14 padded to 16, 4 WMMA steps of K=4.
// A layout: lane m=L%16; VGPR0/1 = K {0,1} (lanes 0-15) / {2,3} (lanes 16-31).
// B layout mirrored: n=L%16, same K striping. D: c[v] -> row v+8*(L>=16), col n.
// Fragments are loaded directly per K-step as aligned float2 (all fragment
// element offsets are even), so no dynamic register indexing is needed; only
// the K=14..15 zero-pad fragment (kb==3, kh==1) is masked with cndmask.
// Reads approx from the out tail (written by kernel 1) and overwrites in place.
// ---------------------------------------------------------------------------
__global__ __launch_bounds__(256) void k_linear_wmma(const float* __restrict__ W,
                                                     const float* __restrict__ bias,
                                                     float* __restrict__ out) {
    const int lane = threadIdx.x & 31;
    const int wave = threadIdx.x >> 5;
    const int tile = blockIdx.x * 8 + wave;   // 16 rows per tile, 1024 tiles
    const int row0 = tile * 16;
    const int m    = lane & 15;               // A-row / B-col / D-col index
    const int kh   = lane >> 4;               // K half select (0 or 1)

    // Per-lane fragment base pointers (element offsets all even -> 8B aligned)
    const float* __restrict__ ar = out + (size_t)(row0 + m) * ROW_OUT + ROW_IN;
    const float* __restrict__ wr = W + m * 14;           // row n = m of W
    const float* ap = ar + 2 * kh;
    const float* wp = wr + 2 * kh;

    v8f c = {};
    #pragma unroll
    for (int kb = 0; kb < 4; ++kb) {
        v2f a, b;
        if (kb < 3) {
            // k0 = 4*kb + 2*kh, max index 11 < 14: always valid
            a = *(const v2f*)(ap + 4 * kb);
            b = *(const v2f*)(wp + 4 * kb);
        } else {
            // kh==0 -> K=12,13 (valid); kh==1 -> K=14,15 (zero pad).
            // Load the valid K=12 fragment on all lanes, mask the kh==1 half.
            v2f at = *(const v2f*)(ar + 12);
            v2f wt = *(const v2f*)(wr + 12);
            const bool hi = (kh != 0);
            a.x = hi ? 0.0f : at.x;  a.y = hi ? 0.0f : at.y;
            b.x = hi ? 0.0f : wt.x;  b.y = hi ? 0.0f : wt.y;
        }
        // 8-arg pattern: (neg_a, A, neg_b, B, c_mod, C, reuse_a, reuse_b)
        c = __builtin_amdgcn_wmma_f32_16x16x4_f32(false, a, false, b,
                                                  (short)0, c, false, false);
    }

    const float bn = (m < 14) ? bias[m] : 0.0f;
    #pragma unroll
    for (int v = 0; v < 8; ++v) {
        const int mo = v + 8 * kh;            // D row within tile
        if (m < 14)
            out[(size_t)(row0 + mo) * ROW_OUT + ROW_IN + m] = c[v] + bn;
    }
}

// ---------------------------------------------------------------------------
extern "C" void kernel_launch(void* const* d_in, const int* in_sizes, int n_in,
                              void* d_out, int out_size, void* d_ws, size_t ws_size,
                              hipStream_t stream) {
    const float* x  = (const float*)d_in[0];   // [32,512,4096]
    const float* W  = (const float*)d_in[1];   // [14,14]
    const float* b  = (const float*)d_in[2];   // [14]
    float* out = (float*)d_out;                // [32,512,4110]

    k_copy_dwt   <<<NROWS, 256, 0, stream>>>(x, out);
    k_linear_wmma<<<NROWS / 16 / 8, 256, 0, stream>>>(W, b, out);
}